// TripletLoss3D_15917148799620
// MI455X (gfx1250) — compile-verified
//
#include <hip/hip_runtime.h>
#include <stdint.h>

typedef float v2f __attribute__((ext_vector_type(2)));
typedef float v8f __attribute__((ext_vector_type(8)));
typedef int   v4i __attribute__((vector_size(4 * sizeof(int))));   // matches builtin param

#define N_ANCHORS 8192
#define DIMK      512
#define BM        32              // anchors per workgroup
#define BN        64              // columns per j-step
#define KC        128             // K chunk staged in LDS
#define A_STRIDE  (DIMK + 4)      // 516 dwords: %64==4 -> conflict-free b64 reads
#define B_STRIDE  (KC + 4)        // 132 dwords: %64==4 -> conflict-free b64 reads
#define JCHUNKS   4
#define JSPAN     (N_ANCHORS / JCHUNKS)            // 2048
#define JSTEPS    (JSPAN / BN)                     // 32
#define TCH       (JSTEPS * (DIMK / KC))           // 128 chunks per WG
#define BLOCK     256
#define MARGIN    1.0f

// gfx1250 async global->LDS path (ASYNCcnt); guarded so the file always compiles.
#if defined(__AMDGCN__) && __has_builtin(__builtin_amdgcn_global_load_async_to_lds_b128) && \
    __has_builtin(__builtin_amdgcn_s_wait_asynccnt)
#define USE_ASYNC 1
#else
#define USE_ASYNC 0
#endif

#if USE_ASYNC
__device__ __forceinline__ void async_cp16(const float* g, float* l) {
  __builtin_amdgcn_global_load_async_to_lds_b128((v4i*)g, (v4i*)l, 0, 0);
}
#endif

struct BTmp { float4 v[8]; };

// Issue the fill of one B chunk (BN x KC): async mode issues LDS-writing DMAs now;
// fallback mode batches the global loads into registers (single wait, no per-load stall).
__device__ __forceinline__ void stageB_issue(BTmp& t, const float* __restrict__ x,
                                             int n0, int kc, float* dst, int tid) {
#if USE_ASYNC
  (void)t;
#pragma unroll
  for (int it = 0; it < (BN * (KC / 4)) / BLOCK; ++it) {   // 8
    int idx = tid + it * BLOCK;
    int r = idx >> 5, c = idx & 31;                         // KC/4 == 32 per row
    async_cp16(x + (size_t)(n0 + r) * DIMK + kc + c * 4, dst + r * B_STRIDE + c * 4);
  }
#else
#pragma unroll
  for (int it = 0; it < 8; ++it) {
    int idx = tid + it * BLOCK;
    int r = idx >> 5, c = idx & 31;
    t.v[it] = *(const float4*)(x + (size_t)(n0 + r) * DIMK + kc + c * 4);
  }
  (void)dst;
#endif
}

// Commit the fill (fallback: LDS stores after the compute phase -> software pipeline).
__device__ __forceinline__ void stageB_commit(BTmp& t, float* dst, int tid) {
#if USE_ASYNC
  (void)t; (void)dst; (void)tid;
#else
#pragma unroll
  for (int it = 0; it < 8; ++it) {
    int idx = tid + it * BLOCK;
    int r = idx >> 5, c = idx & 31;
    *(float4*)(dst + r * B_STRIDE + c * 4) = t.v[it];
  }
#endif
}

// ---------------------------------------------------------------------------
__global__ void tl_init(unsigned* __restrict__ posb, unsigned* __restrict__ negb) {
  int i = blockIdx.x * blockDim.x + threadIdx.x;
  if (i < N_ANCHORS) {
    posb[i] = 0u;                              // dist_pos >= 0 always (diagonal)
    negb[i] = __float_as_uint(1e30f);
  }
}

// ---------------------------------------------------------------------------
__global__ void tl_rownorm(const float* __restrict__ x, float* __restrict__ sq) {
  const int wave = threadIdx.x >> 5;
  const int lane = threadIdx.x & 31;
  const int row  = blockIdx.x * 8 + wave;
  const float* p = x + (size_t)row * DIMK + lane;
  float s = 0.0f;
#pragma unroll
  for (int t = 0; t < DIMK / 32; ++t) {
    float v = p[t * 32];
    s = fmaf(v, v, s);
  }
  for (int m = 16; m; m >>= 1) s += __shfl_xor(s, m, 32);
  if (lane == 0) sq[row] = s;
}

// ---------------------------------------------------------------------------
// Fused streaming Gram kernel: V_WMMA_F32_16X16X4_F32 accumulation, double-
// buffered async B staging, masked running max/min, uint-bit atomics.
// ---------------------------------------------------------------------------
__global__ void __launch_bounds__(BLOCK)
tl_main(const float* __restrict__ x,
        const int*   __restrict__ y2,    // int64 labels viewed as int pairs
        const float* __restrict__ sq,
        unsigned*    __restrict__ posb,
        unsigned*    __restrict__ negb) {
  extern __shared__ float smem[];
  float* As  = smem;                          // BM x A_STRIDE
  float* Bs0 = smem + BM * A_STRIDE;          // BN x B_STRIDE
  float* Bs1 = Bs0 + BN * B_STRIDE;           // BN x B_STRIDE

  const int tid  = threadIdx.x;
  const int wave = tid >> 5;
  const int lane = tid & 31;
  const int ln15 = lane & 15;
  const int hi   = lane >> 4;
  const int wm   = wave >> 2;                 // 0..1
  const int wn   = wave & 3;                  // 0..3
  const int m0   = blockIdx.x * BM;
  const int j0   = blockIdx.y * JSPAN;

  // ---- stage A panel (BM x DIMK), resident for whole kernel ----
#if USE_ASYNC
#pragma unroll
  for (int it = 0; it < (BM * (DIMK / 4)) / BLOCK; ++it) {   // 16
    int idx = tid + it * BLOCK;
    int r = idx >> 7, c = idx & 127;                          // DIMK/4 == 128
    async_cp16(x + (size_t)(m0 + r) * DIMK + c * 4, As + r * A_STRIDE + c * 4);
  }
#else
  for (int half = 0; half < 2; ++half) {
    float4 ta[8];
#pragma unroll
    for (int it = 0; it < 8; ++it) {
      int idx = tid + (half * 8 + it) * BLOCK;
      int r = idx >> 7, c = idx & 127;
      ta[it] = *(const float4*)(x + (size_t)(m0 + r) * DIMK + c * 4);
    }
#pragma unroll
    for (int it = 0; it < 8; ++it) {
      int idx = tid + (half * 8 + it) * BLOCK;
      int r = idx >> 7, c = idx & 127;
      *(float4*)(As + r * A_STRIDE + c * 4) = ta[it];
    }
  }
#endif

  // per-lane anchor metadata (row m = m0 + wm*16 + v + hi*8, matching C layout)
  float sqa[8]; int ya[8];
#pragma unroll
  for (int v = 0; v < 8; ++v) {
    int m  = m0 + wm * 16 + v + hi * 8;
    sqa[v] = sq[m];
    ya[v]  = y2[2 * m];
  }

  float pmax[8], nmin[8];
#pragma unroll
  for (int v = 0; v < 8; ++v) { pmax[v] = 0.0f; nmin[v] = 1e30f; }

  const int aOff = (wm * 16 + ln15) * A_STRIDE + hi * 2;
  const int bOff = (wn * 16 + ln15) * B_STRIDE + hi * 2;

  // ---- prologue: fill chunk 0 into Bs0 ----
  BTmp tmpB;
  stageB_issue(tmpB, x, j0, 0, Bs0, tid);
  stageB_commit(tmpB, Bs0, tid);
#if USE_ASYNC
  __builtin_amdgcn_s_wait_asynccnt(0);
#endif
  __syncthreads();

  v8f acc = {};
  for (int t = 0; t < TCH; ++t) {
    const int js = t >> 2;
    float* cur  = (t & 1) ? Bs1 : Bs0;
    float* next = (t & 1) ? Bs0 : Bs1;
    const bool more = (t + 1) < TCH;

    // issue fill of chunk t+1 (overlaps with compute below)
    if (more) {
      int tn = t + 1;
      stageB_issue(tmpB, x, j0 + (tn >> 2) * BN, (tn & 3) * KC, next, tid);
    }

    // compute: 32 x V_WMMA_F32_16X16X4_F32 on the current chunk
    const float* pa = As + aOff + (t & 3) * KC;
    const float* pb = cur + bOff;
#pragma unroll 8
    for (int kk = 0; kk < KC; kk += 4) {
      v2f a = *(const v2f*)(pa + kk);     // ds_load_b64
      v2f b = *(const v2f*)(pb + kk);     // ds_load_b64
      acc = __builtin_amdgcn_wmma_f32_16x16x4_f32(
          false, a, false, b, (short)0, acc, false, false);
    }

    if (more) stageB_commit(tmpB, next, tid);   // fallback path only
#if USE_ASYNC
    __builtin_amdgcn_s_wait_asynccnt(0);
#endif
    __syncthreads();

    // epilogue once per j-step (after the 4th K chunk)
    if ((t & 3) == 3) {
      const int n0 = j0 + js * BN;
      const int n  = n0 + wn * 16 + ln15;
      const float sqb = sq[n];
      const int   yb  = y2[2 * n];
#pragma unroll
      for (int v = 0; v < 8; ++v) {
        float d = fmaf(-2.0f, acc[v], sqa[v] + sqb);
        d = fmaxf(d, 0.0f);
        if (ya[v] == yb) pmax[v] = fmaxf(pmax[v], d);
        else             nmin[v] = fminf(nmin[v], d);
      }
      acc = (v8f){};
    }
  }

  // reduce each row across its 16-lane half-group, then combine globally
#pragma unroll
  for (int v = 0; v < 8; ++v) {
    float p = pmax[v], q = nmin[v];
    for (int s = 1; s < 16; s <<= 1) {
      p = fmaxf(p, __shfl_xor(p, s, 32));
      q = fminf(q, __shfl_xor(q, s, 32));
    }
    if (ln15 == 0) {
      int m = m0 + wm * 16 + v + hi * 8;
      atomicMax(&posb[m], __float_as_uint(p));
      atomicMin(&negb[m], __float_as_uint(q));
    }
  }
}

// ---------------------------------------------------------------------------
__global__ void tl_finalize(const unsigned* __restrict__ posb,
                            const unsigned* __restrict__ negb,
                            float* __restrict__ out) {
  __shared__ float red[256];
  float s = 0.0f;
  for (int i = threadIdx.x; i < N_ANCHORS; i += 256) {
    float p = __uint_as_float(posb[i]);
    float q = __uint_as_float(negb[i]);
    s += fmaxf(p + MARGIN - q, 0.0f);
  }
  red[threadIdx.x] = s;
  __syncthreads();
  for (int off = 128; off; off >>= 1) {
    if (threadIdx.x < off) red[threadIdx.x] += red[threadIdx.x + off];
    __syncthreads();
  }
  if (threadIdx.x == 0) out[0] = red[0] / (float)N_ANCHORS;
}

// ---------------------------------------------------------------------------
extern "C" void kernel_launch(void* const* d_in, const int* in_sizes, int n_in,
                              void* d_out, int out_size, void* d_ws, size_t ws_size,
                              hipStream_t stream) {
  (void)in_sizes; (void)n_in; (void)out_size; (void)ws_size;
  const float* x  = (const float*)d_in[0];
  const int*   y2 = (const int*)d_in[1];           // int64 labels, read low dwords

  float*    sq   = (float*)d_ws;                                    // 8192 f32
  unsigned* posb = (unsigned*)((char*)d_ws + N_ANCHORS * sizeof(float));
  unsigned* negb = posb + N_ANCHORS;

  tl_init<<<N_ANCHORS / 256, 256, 0, stream>>>(posb, negb);
  tl_rownorm<<<N_ANCHORS / 8, 256, 0, stream>>>(x, sq);

  dim3 grid(N_ANCHORS / BM, JCHUNKS);
  size_t smem_bytes = (size_t)(BM * A_STRIDE + 2 * BN * B_STRIDE) * sizeof(float);
  tl_main<<<grid, BLOCK, smem_bytes, stream>>>(x, y2, sq, posb, negb);

  tl_finalize<<<1, 256, 0, stream>>>(posb, negb, (float*)d_out);
}